// BeBertEmbedding_68899865362566
// MI455X (gfx1250) — compile-verified
//
#include <hip/hip_runtime.h>

typedef __attribute__((ext_vector_type(2))) float v2f;
typedef __attribute__((ext_vector_type(8))) float v8f;

constexpr int D     = 768;
constexpr int S     = 2048;
constexpr int VOCAB = 32000;
constexpr int SEGV  = 3;

// out[b,s,d] = pe[s,d] + f_tok[b,s]*p_tok[d] + f_seg[b,s]*p_seg[d]
//   f_*[b,s] = (index == 0) ? 1 : 0          (only pad tokens keep emb,
//                                             and pad gathers column 0)
//   p_tok[d] = W_tok[d,0] + b_tok[d],  p_seg[d] = W_seg[d,0] + b_seg[d]
//
// One wave computes one 16(batch) x 16(d) fp32 output tile at fixed s via
// V_WMMA_F32_16X16X4_F32:  D = A(16x4) * B(4x16) + 0
//   A rows (M=b): [f_tok, f_seg, 1, 0]
//   B rows (K):   [p_tok[d], p_seg[d], pe[s,d], 0]
__global__ __launch_bounds__(256)
void embed_wmma(const int* __restrict__ ids,
                const int* __restrict__ segs,
                const float* __restrict__ W_tok,
                const float* __restrict__ b_tok,
                const float* __restrict__ W_seg,
                const float* __restrict__ b_seg,
                const float* __restrict__ pe,
                float* __restrict__ out) {
  const int s    = blockIdx.x;                  // 0..2047
  const int wave = threadIdx.x >> 5;            // 0..7
  const int lane = threadIdx.x & 31;
  const int tile = blockIdx.y * 8 + wave;       // 0..47
  const int d0   = tile * 16;
  const int n    = lane & 15;
  const int d    = d0 + n;
  const bool hi  = lane >= 16;

  // Branchless loads; addresses valid for every lane, values selected below.
  const float ftok = (ids [n * S + s] == 0) ? 1.0f : 0.0f;
  const float fseg = (segs[n * S + s] == 0) ? 1.0f : 0.0f;
  // p vectors computed in-kernel: 768 distinct values, L2-resident after
  // first touch, so no prologue kernel / graph node is needed.
  const float ptok = W_tok[(size_t)d * VOCAB] + b_tok[d];
  const float pseg = W_seg[d * SEGV] + b_seg[d];
  const float pev  = pe[(size_t)s * D + d];

  // A-matrix 16x4 f32 layout: VGPR0 = K0 (lanes 0-15) / K2 (lanes 16-31),
  //                           VGPR1 = K1 (lanes 0-15) / K3 (lanes 16-31).
  v2f a, b;
  a.x = hi ? 1.0f : ftok;   // K=0: f_tok      | K=2: 1 (pe coefficient)
  a.y = hi ? 0.0f : fseg;   // K=1: f_seg      | K=3: 0
  // B-matrix 4x16 f32, row striped across lanes (mirror of A split):
  b.x = hi ? pev  : ptok;   // K=0 row: p_tok  | K=2 row: pe[s,d]
  b.y = hi ? 0.0f : pseg;   // K=1 row: p_seg  | K=3 row: 0

  v8f c = {};
  c = __builtin_amdgcn_wmma_f32_16x16x4_f32(
      /*neg_a=*/false, a, /*neg_b=*/false, b,
      /*c_mod=*/(short)0, c, /*reuse_a=*/false, /*reuse_b=*/false);

  // D layout: VGPR r holds row M=r (lanes 0-15) / M=r+8 (lanes 16-31), N=n.
  const int m0 = hi ? 8 : 0;
  float* o = out + ((size_t)(m0 * S + s)) * D + d;
#pragma unroll
  for (int r = 0; r < 8; ++r) {
    o[(size_t)r * S * D] = c[r];
  }
}

extern "C" void kernel_launch(void* const* d_in, const int* in_sizes, int n_in,
                              void* d_out, int out_size, void* d_ws, size_t ws_size,
                              hipStream_t stream) {
  const int*   ids   = (const int*)  d_in[0];
  const int*   segs  = (const int*)  d_in[1];
  const float* W_tok = (const float*)d_in[2];
  const float* b_tok = (const float*)d_in[3];
  const float* W_seg = (const float*)d_in[4];
  const float* b_seg = (const float*)d_in[5];
  const float* pe    = (const float*)d_in[6];
  float* out = (float*)d_out;

  dim3 grid(S, (D / 16) / 8);  // (2048, 6): 8 waves/block -> 8 d-tiles
  embed_wmma<<<grid, 256, 0, stream>>>(ids, segs, W_tok, b_tok, W_seg, b_seg,
                                       pe, out);
}